// SparseNeuronRetriever_67216238182667
// MI455X (gfx1250) — compile-verified
//
#include <hip/hip_runtime.h>
#include <hip/hip_bf16.h>
#include <cfloat>
#include <cstdint>

#define EPS 1e-12f

typedef __attribute__((ext_vector_type(2))) float v2f;
typedef __attribute__((ext_vector_type(8))) float v8f;

// ---------------------------------------------------------------------------
// Kernel 0: ws0[0] = 1 / max(||q||, eps)
// ---------------------------------------------------------------------------
__global__ __launch_bounds__(256) void qnorm_kernel(const float* __restrict__ q,
                                                    int D,
                                                    float* __restrict__ ws0) {
    __shared__ float red[256];
    float s = 0.f;
    for (int i = threadIdx.x * 4; i + 3 < D; i += 1024) {
        const float4 v = *(const float4*)(q + i);
        s = fmaf(v.x, v.x, s);
        s = fmaf(v.y, v.y, s);
        s = fmaf(v.z, v.z, s);
        s = fmaf(v.w, v.w, s);
    }
    red[threadIdx.x] = s;
    __syncthreads();
    for (int off = 128; off > 0; off >>= 1) {
        if (threadIdx.x < (unsigned)off) red[threadIdx.x] += red[threadIdx.x + off];
        __syncthreads();
    }
    if (threadIdx.x == 0) ws0[0] = 1.0f / fmaxf(sqrtf(red[0]), EPS);
}

// ---------------------------------------------------------------------------
// Kernel 1: sims[row] = dot(C[row], q) / (max(||C[row]||,eps) * max(||q||,eps))
// 8 waves / block, 16 rows / wave -> 128 rows / block.
// K-step of 8 per lane via one 128-bit load; two WMMA f32 16x16x4 per step
// over interleaved K sets so the float4 lines up with the A-matrix layout:
//   wmma0 <-> K {kk, kk+1, kk+4, kk+5}   (lo lanes t.xy, hi lanes t.xy)
//   wmma1 <-> K {kk+2, kk+3, kk+6, kk+7} (lo lanes t.zw, hi lanes t.zw)
// ---------------------------------------------------------------------------
__global__ __launch_bounds__(256) void gemv_wmma_kernel(
    const float* __restrict__ q, const float* __restrict__ chunks,
    const float* __restrict__ ws0, float* __restrict__ sims,
    int n_rows, int D) {
    __shared__ __align__(16) float qs[2048];
    __shared__ float s_dot[128];
    __shared__ float s_ns[128];

    const int tid = threadIdx.x;
    for (int i = tid * 4; i + 3 < D && i + 3 < 2048; i += 1024)
        *(float4*)(qs + i) = *(const float4*)(q + i);
    __syncthreads();

    const int lane = tid & 31;
    const int wv   = tid >> 5;
    const int m    = lane & 15;            // row within the wave's 16-row tile
    const int koff = (lane >> 4) << 2;     // 0 for lanes 0-15, 4 for lanes 16-31

    int row  = blockIdx.x * 128 + wv * 16 + m;
    int lrow = (row < n_rows) ? row : (n_rows - 1);   // clamp: keep EXEC all-1s for WMMA
    const float* rp = chunks + (size_t)lrow * (size_t)D + koff;
    const float* qp = qs + koff;

    const float bmask = (m == 0) ? 1.f : 0.f;  // only N=0 column of B is live

    v8f   acc = {};
    float ns  = 0.f;

#pragma unroll 2
    for (int kk = 0; kk < D; kk += 8) {
        const float4 t  = *(const float4*)(rp + kk);  // global_load_b128
        const float4 qv = *(const float4*)(qp + kk);  // ds_load_b128 (half-wave broadcast)

        v2f a0; a0.x = t.x; a0.y = t.y;
        v2f a1; a1.x = t.z; a1.y = t.w;
        v2f b0; b0.x = qv.x * bmask; b0.y = qv.y * bmask;
        v2f b1; b1.x = qv.z * bmask; b1.y = qv.w * bmask;

        ns = fmaf(t.x, t.x, ns);
        ns = fmaf(t.y, t.y, ns);
        ns = fmaf(t.z, t.z, ns);
        ns = fmaf(t.w, t.w, ns);

        acc = __builtin_amdgcn_wmma_f32_16x16x4_f32(false, a0, false, b0,
                                                    (short)0, acc, false, false);
        acc = __builtin_amdgcn_wmma_f32_16x16x4_f32(false, a1, false, b1,
                                                    (short)0, acc, false, false);
    }

    // combine normsq halves: lane L covers K offsets {0..3}+koff of each 8-group;
    // partner lane L^16 covers the other 4.
    ns += __shfl_xor(ns, 16, 32);

    // D[:,0] lives in lane 0 (M=0..7 -> acc[0..7]) and lane 16 (M=8..15 -> acc[0..7])
    if (lane == 0) {
#pragma unroll
        for (int i = 0; i < 8; ++i) s_dot[wv * 16 + i] = acc[i];
    }
    if (lane == 16) {
#pragma unroll
        for (int i = 0; i < 8; ++i) s_dot[wv * 16 + 8 + i] = acc[i];
    }
    if (lane < 16) s_ns[wv * 16 + lane] = ns;
    __syncthreads();

    if (tid < 128) {
        const int orow = blockIdx.x * 128 + tid;
        if (orow < n_rows) {
            const float cn = fmaxf(sqrtf(s_ns[tid]), EPS);
            sims[orow] = (s_dot[tid] / cn) * ws0[0];
        }
    }
}

// ---------------------------------------------------------------------------
// Kernel 2: iterative top-k (k rounds of block-wide argmax over n sims).
// Ties break toward the lower index (matches lax.top_k). Winner masked to
// -FLT_MAX; sims is fully rewritten by kernel 1 each call -> deterministic.
// Output layout: out[0..k-1] = indices (as float), out[k..2k-1] = scores.
// ---------------------------------------------------------------------------
__global__ __launch_bounds__(1024) void topk_kernel(float* __restrict__ sims,
                                                    float* __restrict__ out,
                                                    int n, int k) {
    __shared__ float bv[1024];
    __shared__ int   bi[1024];
    const int tid = threadIdx.x;

    for (int sel = 0; sel < k; ++sel) {
        float best = -FLT_MAX;
        int   bidx = 0x7FFFFFFF;
        for (int i = tid; i < n; i += 1024) {
            const float v = sims[i];
            if (v > best || (v == best && i < bidx)) { best = v; bidx = i; }
        }
        bv[tid] = best;
        bi[tid] = bidx;
        __syncthreads();
        for (int off = 512; off > 0; off >>= 1) {
            if (tid < off) {
                const float v2 = bv[tid + off];
                const int   i2 = bi[tid + off];
                if (v2 > bv[tid] || (v2 == bv[tid] && i2 < bi[tid])) {
                    bv[tid] = v2;
                    bi[tid] = i2;
                }
            }
            __syncthreads();
        }
        if (tid == 0) {
            const int widx = bi[0];
            out[sel]     = (float)widx;
            out[k + sel] = bv[0];
            sims[widx]   = -FLT_MAX;  // mask winner for next round
        }
        __threadfence();
        __syncthreads();
    }
}

// ---------------------------------------------------------------------------
extern "C" void kernel_launch(void* const* d_in, const int* in_sizes, int n_in,
                              void* d_out, int out_size, void* d_ws, size_t ws_size,
                              hipStream_t stream) {
    const float* q      = (const float*)d_in[0];
    const float* chunks = (const float*)d_in[1];
    float*       out    = (float*)d_out;

    const int D      = in_sizes[0];          // 2048
    const int n_rows = in_sizes[1] / D;      // 100000
    int k = out_size / 2;                    // (indices, scores) concatenated
    if (k > n_rows) k = n_rows;

    float* ws0  = (float*)d_ws;              // [0]: 1/max(||q||,eps)
    float* sims = (float*)d_ws + 64;         // 256 B offset, then n_rows floats

    qnorm_kernel<<<1, 256, 0, stream>>>(q, D, ws0);

    const int blocks = (n_rows + 127) / 128;
    gemv_wmma_kernel<<<blocks, 256, 0, stream>>>(q, chunks, ws0, sims, n_rows, D);

    topk_kernel<<<1, 1024, 0, stream>>>(sims, out, n_rows, k);
}